// VSLNet_76459007803459
// MI455X (gfx1250) — compile-verified
//
#include <hip/hip_runtime.h>
#include <hip/hip_bf16.h>

// ---------------- problem constants ----------------
#define B_    8
#define T_    256
#define LQ_   20
#define LC_   10
#define DIM_  128
#define H_    8
#define HD_   16
#define VDIM_ 1024
#define WDIM_ 300
#define CDIM_ 50
#define N_    257            // T+1 nodes per batch
#define NN_   2056           // B*N
#define NNP_  2064           // padded to 16
#define E1_   67067          // edges per batch
#define E_    536536         // total edges (B*E1)

// per-batch edge segment offsets
#define oA_ 0
#define oB_ 255
#define oC_ 510
#define oD_ 764
#define oE_ 1018
#define oF_ 1274
#define oSF_ 1530
#define oSB_ 34170
#define oQ_ 66810

// workspace layout (float-element offsets)
#define WS_VTMP  0                         // 2048*128      = 262144
#define WS_XF    262144                    // 2064*128      = 264192
#define WS_Q     526336
#define WS_K     790528
#define WS_V     1054720
#define WS_SKIP  1318912
#define WS_AGG   1583104
#define WS_DEN   1847296                   // 2064*8 = 16512
#define WS_MX    1863808                   // 16512 (as uint keys)
#define WS_EDGE  1880320                   // 2*E ints = 1073072

typedef float v2f __attribute__((ext_vector_type(2)));
typedef float v8f __attribute__((ext_vector_type(8)));

#if defined(__HIP_DEVICE_COMPILE__)
#if !__has_builtin(__builtin_amdgcn_wmma_f32_16x16x4_f32)
#error "CDNA5 device pass: __builtin_amdgcn_wmma_f32_16x16x4_f32 not available"
#endif
#endif

__device__ __forceinline__ v8f wmma4(v2f a, v2f b, v8f c) {
#if defined(__HIP_DEVICE_COMPILE__)
  // v_wmma_f32_16x16x4_f32: D = A(16x4 f32) * B(4x16 f32) + C(16x16 f32)
  return __builtin_amdgcn_wmma_f32_16x16x4_f32(false, a, false, b, (short)0, c,
                                               false, false);
#else
  (void)a; (void)b;
  return c;  // host pass: parsed but never executed
#endif
}

// one wave computes one 16x16 f32 tile: C = A[row0:row0+16, 0:K] * W[0:K, col0:col0+16]
// A lda arbitrary, W is row-major with leading dim DIM_ (128).
__device__ __forceinline__ v8f wave_tile_gemm(const float* __restrict__ A, int lda,
                                              const float* __restrict__ W,
                                              int row0, int col0, int K) {
  const int lane = threadIdx.x & 31;
  const int half = lane >> 4;   // selects K pair {0,1} vs {2,3}
  const int l16  = lane & 15;   // A: M row / B: N col
  const float* arow = A + (size_t)(row0 + l16) * lda;
  const float* wcol = W + col0 + l16;
  v8f acc = {};
  for (int kk = 0; kk < K; kk += 4) {
    const int ka = kk + 2 * half;
    v2f a, b;
    a.x = arow[ka];
    a.y = arow[ka + 1];
    b.x = wcol[(size_t)ka * DIM_];
    b.y = wcol[(size_t)(ka + 1) * DIM_];
    acc = wmma4(a, b, acc);
  }
  return acc;
}

// ---------------- init ----------------
__global__ void __launch_bounds__(256) init_kernel(float* agg, float* den,
                                                   unsigned* mxU, float* xf) {
  const int g = blockIdx.x * 256 + threadIdx.x;
  if (g < NNP_ * DIM_) agg[g] = 0.0f;
  if (g < NNP_ * H_) { den[g] = 0.0f; mxU[g] = 0u; }  // key 0 == minimum
  if (g < (NNP_ - NN_) * DIM_) xf[(size_t)NN_ * DIM_ + g] = 0.0f;  // pad rows
}

// ---------------- edge builders ----------------
__global__ void __launch_bounds__(256) build_edges_simple(int* edges) {
  const int g = blockIdx.x * 256 + threadIdx.x;
  if (g >= B_ * N_) return;
  const int b = g / N_, i = g % N_;
  const int base = b * E1_, nb = b * N_;
  if (i < T_ - 1) {  // hop 1
    int e = base + oA_ + i; edges[2 * e] = nb + i;     edges[2 * e + 1] = nb + i + 1;
    e     = base + oB_ + i; edges[2 * e] = nb + i + 1; edges[2 * e + 1] = nb + i;
  }
  if (i < T_ - 2) {  // hop 2
    int e = base + oC_ + i; edges[2 * e] = nb + i;     edges[2 * e + 1] = nb + i + 2;
    e     = base + oD_ + i; edges[2 * e] = nb + i + 2; edges[2 * e + 1] = nb + i;
  }
  if (i < T_) {      // query<->segment
    int e = base + oE_ + i; edges[2 * e] = nb;         edges[2 * e + 1] = nb + i + 1;
    e     = base + oF_ + i; edges[2 * e] = nb + i + 1; edges[2 * e + 1] = nb;
  }
  {                  // query cycle edges
    int e = base + oQ_ + i; edges[2 * e] = nb + i;     edges[2 * e + 1] = nb + (i + 1) % N_;
  }
}

__global__ void __launch_bounds__(256) build_edges_sem(int* edges) {
  const int g = blockIdx.x * 256 + threadIdx.x;  // B * 256 * 256 threads
  const int b = g >> 16, r = g & 0xFFFF;
  const int a = r >> 8, c = r & 0xFF;
  if (b >= B_ || a >= c) return;
  const int t = (T_ - 1) * a - (a * (a - 1)) / 2 + (c - a - 1);  // exact triu index
  const int base = b * E1_, nb = b * N_;
  int e = base + oSF_ + t; edges[2 * e] = nb + a; edges[2 * e + 1] = nb + c;
  e     = base + oSB_ + t; edges[2 * e] = nb + c; edges[2 * e + 1] = nb + a;
}

// ---------------- query branch ----------------
__global__ void __launch_bounds__(128) query_kernel(
    const int* __restrict__ word_ids, const int* __restrict__ char_ids,
    const float* __restrict__ q_mask, const float* __restrict__ word_emb,
    const float* __restrict__ char_emb, const float* __restrict__ W_embed,
    const float* __restrict__ b_embed, const float* __restrict__ W_enc,
    const float* __restrict__ b_enc, float* __restrict__ xf) {
  __shared__ float qemb[LQ_][WDIM_ + CDIM_ + 2];  // 352-wide to dodge conflicts
  __shared__ float h1[LQ_][DIM_];
  const int b = blockIdx.x, tid = threadIdx.x;
  for (int idx = tid; idx < LQ_ * WDIM_; idx += 128) {
    const int l = idx / WDIM_, j = idx % WDIM_;
    qemb[l][j] = word_emb[(size_t)word_ids[b * LQ_ + l] * WDIM_ + j];
  }
  for (int idx = tid; idx < LQ_ * CDIM_; idx += 128) {
    const int l = idx / CDIM_, j = idx % CDIM_;
    float s = 0.0f;
    for (int c = 0; c < LC_; ++c)
      s += char_emb[(size_t)char_ids[(b * LQ_ + l) * LC_ + c] * CDIM_ + j];
    qemb[l][WDIM_ + j] = s * (1.0f / LC_);
  }
  __syncthreads();
  const int d = tid;  // output dim
  for (int l = 0; l < LQ_; ++l) {
    float s = b_embed[d];
    for (int kk = 0; kk < WDIM_ + CDIM_; ++kk)
      s += qemb[l][kk] * W_embed[kk * DIM_ + d];
    h1[l][d] = s;
  }
  __syncthreads();
  float acc = 0.0f;
  for (int l = 0; l < LQ_; ++l) {
    float s = b_enc[d];
    for (int kk = 0; kk < DIM_; ++kk) s += h1[l][kk] * W_enc[kk * DIM_ + d];
    acc += fmaxf(s, 0.0f) * q_mask[b * LQ_ + l];
  }
  xf[(size_t)(b * N_) * DIM_ + d] = acc * (1.0f / LQ_);  // node 0 of batch b
}

// ---------------- WMMA GEMMs ----------------
// vtmp = video @ W_vproj + b_vproj   (M=2048, K=1024, N=128)
__global__ void __launch_bounds__(256) gemm_vproj(const float* __restrict__ A,
                                                  const float* __restrict__ W,
                                                  const float* __restrict__ bias,
                                                  float* __restrict__ out) {
  const int lane = threadIdx.x & 31, half = lane >> 4, l16 = lane & 15;
  const int row0 = blockIdx.x * 16, col0 = (threadIdx.x >> 5) * 16;
  v8f acc = wave_tile_gemm(A, VDIM_, W, row0, col0, VDIM_);
  const float bs = bias[col0 + l16];
  for (int r = 0; r < 8; ++r)
    out[(size_t)(row0 + r + 8 * half) * DIM_ + col0 + l16] = acc[r] + bs;
}

// xf[b*257+1+t] = relu(vtmp @ W_enc + b_enc) * v_mask   (M=2048, K=128, N=128)
__global__ void __launch_bounds__(256) gemm_enc(const float* __restrict__ A,
                                                const float* __restrict__ W,
                                                const float* __restrict__ bias,
                                                const float* __restrict__ v_mask,
                                                float* __restrict__ xf) {
  const int lane = threadIdx.x & 31, half = lane >> 4, l16 = lane & 15;
  const int row0 = blockIdx.x * 16, col0 = (threadIdx.x >> 5) * 16;
  v8f acc = wave_tile_gemm(A, DIM_, W, row0, col0, DIM_);
  const float bs = bias[col0 + l16];
  for (int r = 0; r < 8; ++r) {
    const int m = row0 + r + 8 * half;          // (b,t) flat
    const int bb = m >> 8, tt = m & 255;
    const float val = fmaxf(acc[r] + bs, 0.0f) * v_mask[m];
    xf[(size_t)(bb * N_ + 1 + tt) * DIM_ + col0 + l16] = val;
  }
}

// q/k/v/skip = xf @ W* + b*   (M=2064 padded, K=128, N=128) x 4 matrices
__global__ void __launch_bounds__(256) gemm_qkvs(
    const float* __restrict__ xf, const float* __restrict__ Wq,
    const float* __restrict__ bq, const float* __restrict__ Wk,
    const float* __restrict__ bk, const float* __restrict__ Wv,
    const float* __restrict__ bv, const float* __restrict__ Ws,
    const float* __restrict__ bs, float* __restrict__ oq, float* __restrict__ ok,
    float* __restrict__ ov, float* __restrict__ os) {
  const int lane = threadIdx.x & 31, half = lane >> 4, l16 = lane & 15;
  const int row0 = blockIdx.x * 16, col0 = (threadIdx.x >> 5) * 16;
  const float* W; const float* bias; float* out;
  switch (blockIdx.y) {
    case 0:  W = Wq; bias = bq; out = oq; break;
    case 1:  W = Wk; bias = bk; out = ok; break;
    case 2:  W = Wv; bias = bv; out = ov; break;
    default: W = Ws; bias = bs; out = os; break;
  }
  v8f acc = wave_tile_gemm(xf, DIM_, W, row0, col0, DIM_);
  const float bb = bias[col0 + l16];
  for (int r = 0; r < 8; ++r)
    out[(size_t)(row0 + r + 8 * half) * DIM_ + col0 + l16] = acc[r] + bb;
}

// ---------------- edge attention (segment softmax via atomics) ----------------
__device__ __forceinline__ unsigned fkey(float f) {  // monotone float->uint
  unsigned b = __float_as_uint(f);
  return (b & 0x80000000u) ? ~b : (b | 0x80000000u);
}
__device__ __forceinline__ float funkey(unsigned k) {
  return __uint_as_float((k & 0x80000000u) ? (k ^ 0x80000000u) : ~k);
}
__device__ __forceinline__ void edge_logits(const float* __restrict__ q,
                                            const float* __restrict__ k, int s,
                                            int d, float* lg) {
  const float* qr = q + (size_t)d * DIM_;
  const float* kr = k + (size_t)s * DIM_;
#pragma unroll
  for (int h = 0; h < H_; ++h) {
    float acc = 0.0f;
#pragma unroll
    for (int j = 0; j < HD_; ++j) acc += qr[h * HD_ + j] * kr[h * HD_ + j];
    lg[h] = acc * 0.25f;  // 1/sqrt(16)
  }
}

__global__ void __launch_bounds__(256) attn_max(const int* __restrict__ edges,
                                                const float* __restrict__ q,
                                                const float* __restrict__ k,
                                                unsigned* __restrict__ mxU) {
  const int e = blockIdx.x * 256 + threadIdx.x;
  if (e >= E_) return;
  const int s = edges[2 * e], d = edges[2 * e + 1];
  float lg[H_];
  edge_logits(q, k, s, d, lg);
#pragma unroll
  for (int h = 0; h < H_; ++h) atomicMax(&mxU[d * H_ + h], fkey(lg[h]));
}

__global__ void __launch_bounds__(256) attn_den(const int* __restrict__ edges,
                                                const float* __restrict__ q,
                                                const float* __restrict__ k,
                                                const unsigned* __restrict__ mxU,
                                                float* __restrict__ den) {
  const int e = blockIdx.x * 256 + threadIdx.x;
  if (e >= E_) return;
  const int s = edges[2 * e], d = edges[2 * e + 1];
  float lg[H_];
  edge_logits(q, k, s, d, lg);
#pragma unroll
  for (int h = 0; h < H_; ++h)
    atomicAdd(&den[d * H_ + h], __expf(lg[h] - funkey(mxU[d * H_ + h])));
}

__global__ void __launch_bounds__(256) attn_agg(const int* __restrict__ edges,
                                                const float* __restrict__ q,
                                                const float* __restrict__ k,
                                                const float* __restrict__ v,
                                                const unsigned* __restrict__ mxU,
                                                const float* __restrict__ den,
                                                float* __restrict__ agg) {
  const int e = blockIdx.x * 256 + threadIdx.x;
  if (e >= E_) return;
  const int s = edges[2 * e], d = edges[2 * e + 1];
  float lg[H_];
  edge_logits(q, k, s, d, lg);
#pragma unroll
  for (int h = 0; h < H_; ++h) {
    const float alpha = __expf(lg[h] - funkey(mxU[d * H_ + h])) /
                        (den[d * H_ + h] + 1e-16f);
#pragma unroll
    for (int j = 0; j < HD_; ++j)
      atomicAdd(&agg[(size_t)d * DIM_ + h * HD_ + j],
                alpha * v[(size_t)s * DIM_ + h * HD_ + j]);
  }
}

// ---------------- final logits ----------------
__global__ void __launch_bounds__(256) final_kernel(
    const float* __restrict__ agg, const float* __restrict__ skip,
    const float* __restrict__ W_start, const float* __restrict__ b_start,
    const float* __restrict__ W_end, const float* __restrict__ b_end,
    float* __restrict__ out) {
  const int g = blockIdx.x * 256 + threadIdx.x;
  if (g >= B_ * T_) return;
  const int b = g >> 8, t = g & 255;
  const size_t row = (size_t)(b * N_ + 1 + t) * DIM_;
  float s = 0.0f, e = 0.0f;
  for (int d = 0; d < DIM_; ++d) {
    const float o = agg[row + d] + skip[row + d];
    s += o * W_start[d];
    e += o * W_end[d];
  }
  out[g] = s + b_start[0];
  out[B_ * T_ + g] = e + b_end[0];
}

// ---------------- launcher ----------------
extern "C" void kernel_launch(void* const* d_in, const int* in_sizes, int n_in,
                              void* d_out, int out_size, void* d_ws,
                              size_t ws_size, hipStream_t stream) {
  const int*   word_ids = (const int*)d_in[0];
  const int*   char_ids = (const int*)d_in[1];
  const float* video    = (const float*)d_in[2];
  const float* v_mask   = (const float*)d_in[3];
  const float* q_mask   = (const float*)d_in[4];
  const float* word_emb = (const float*)d_in[5];
  const float* char_emb = (const float*)d_in[6];
  const float* W_embed  = (const float*)d_in[7];
  const float* b_embed  = (const float*)d_in[8];
  const float* W_vproj  = (const float*)d_in[9];
  const float* b_vproj  = (const float*)d_in[10];
  const float* W_enc    = (const float*)d_in[11];
  const float* b_enc    = (const float*)d_in[12];
  const float* Wq       = (const float*)d_in[13];
  const float* bq       = (const float*)d_in[14];
  const float* Wk       = (const float*)d_in[15];
  const float* bk       = (const float*)d_in[16];
  const float* Wv       = (const float*)d_in[17];
  const float* bv       = (const float*)d_in[18];
  const float* Wskip    = (const float*)d_in[19];
  const float* bskip    = (const float*)d_in[20];
  const float* W_start  = (const float*)d_in[21];
  const float* b_start  = (const float*)d_in[22];
  const float* W_end    = (const float*)d_in[23];
  const float* b_end    = (const float*)d_in[24];

  float*    ws    = (float*)d_ws;
  float*    vtmp  = ws + WS_VTMP;
  float*    xf    = ws + WS_XF;
  float*    qbuf  = ws + WS_Q;
  float*    kbuf  = ws + WS_K;
  float*    vbuf  = ws + WS_V;
  float*    sbuf  = ws + WS_SKIP;
  float*    agg   = ws + WS_AGG;
  float*    den   = ws + WS_DEN;
  unsigned* mxU   = (unsigned*)(ws + WS_MX);
  int*      edges = (int*)(ws + WS_EDGE);
  float*    out   = (float*)d_out;

  init_kernel<<<(NNP_ * DIM_ + 255) / 256, 256, 0, stream>>>(agg, den, mxU, xf);
  build_edges_simple<<<(B_ * N_ + 255) / 256, 256, 0, stream>>>(edges);
  build_edges_sem<<<(B_ * 256 * 256) / 256, 256, 0, stream>>>(edges);
  query_kernel<<<B_, 128, 0, stream>>>(word_ids, char_ids, q_mask, word_emb,
                                       char_emb, W_embed, b_embed, W_enc, b_enc,
                                       xf);
  gemm_vproj<<<(B_ * T_) / 16, 256, 0, stream>>>(video, W_vproj, b_vproj, vtmp);
  gemm_enc<<<(B_ * T_) / 16, 256, 0, stream>>>(vtmp, W_enc, b_enc, v_mask, xf);
  gemm_qkvs<<<dim3(NNP_ / 16, 4), 256, 0, stream>>>(
      xf, Wq, bq, Wk, bk, Wv, bv, Wskip, bskip, qbuf, kbuf, vbuf, sbuf);
  const int eblocks = (E_ + 255) / 256;
  attn_max<<<eblocks, 256, 0, stream>>>(edges, qbuf, kbuf, mxU);
  attn_den<<<eblocks, 256, 0, stream>>>(edges, qbuf, kbuf, mxU, den);
  attn_agg<<<eblocks, 256, 0, stream>>>(edges, qbuf, kbuf, vbuf, mxU, den, agg);
  final_kernel<<<(B_ * T_ + 255) / 256, 256, 0, stream>>>(
      agg, sbuf, W_start, b_start, W_end, b_end, out);
}